// VarianceLoss_82325933130209
// MI455X (gfx1250) — compile-verified
//
#include <hip/hip_runtime.h>
#include <hip/hip_bf16.h>

// VarianceLoss: per-column unbiased variance of a [n, 1024] fp32 matrix,
// loss = sum(|var - 1|). Memory-bound: 256 MB single-pass read -> ~11 us at
// 23.3 TB/s. Stage 1 streams with b128 NT loads + fp32 atomics into 8 KB ws.
// Stage 2 is one wave32 using v_wmma_f32_16x16x4_f32 as a cross-lane reducer.

#define NCOLS 1024
#define C4 (NCOLS / 4)          // 256 float4 column groups
#define ROWS_PER_BLOCK 128

typedef __attribute__((ext_vector_type(2))) float v2f;
typedef __attribute__((ext_vector_type(4))) float v4f;
typedef __attribute__((ext_vector_type(8))) float v8f;

__global__ void vl_zero_ws(float* __restrict__ ws) {
    ws[blockIdx.x * 256 + threadIdx.x] = 0.0f;   // zero 2048 floats with <<<8,256>>>
}

// Stage 1: each block owns ROWS_PER_BLOCK consecutive rows; thread t owns
// columns [4t, 4t+3]. A wave reads 512 contiguous bytes per row step ->
// perfectly coalesced global_load_b128 (non-temporal: one-shot 256 MB stream,
// keep it out of L2). Accumulate S = sum(x), Q = sum(x^2) per column in
// registers, then one atomic f32 add per accumulator.
__global__ void vl_stage1(const float* __restrict__ x,
                          float* __restrict__ ws,
                          int n) {
    const int tid = threadIdx.x;                 // 0..255 -> float4 column group
    const long r0 = (long)blockIdx.x * ROWS_PER_BLOCK;
    const int rlim = (int)min((long)ROWS_PER_BLOCK, (long)n - r0);

    const v4f* __restrict__ xp = (const v4f*)x;  // row stride C4
    long base = r0 * (long)C4 + tid;

    float sx = 0.f, sy = 0.f, sz = 0.f, sw = 0.f;
    float qx = 0.f, qy = 0.f, qz = 0.f, qw = 0.f;

#pragma unroll 4
    for (int r = 0; r < rlim; ++r) {
        v4f v = __builtin_nontemporal_load(&xp[base + (long)r * C4]);
        sx += v.x; sy += v.y; sz += v.z; sw += v.w;
        qx = fmaf(v.x, v.x, qx);
        qy = fmaf(v.y, v.y, qy);
        qz = fmaf(v.z, v.z, qz);
        qw = fmaf(v.w, v.w, qw);
    }

    float* __restrict__ s_ws = ws;               // [0, 1024): column sums
    float* __restrict__ q_ws = ws + NCOLS;       // [1024, 2048): column sumsq
    const int c = tid * 4;
    atomicAdd(&s_ws[c + 0], sx);
    atomicAdd(&s_ws[c + 1], sy);
    atomicAdd(&s_ws[c + 2], sz);
    atomicAdd(&s_ws[c + 3], sw);
    atomicAdd(&q_ws[c + 0], qx);
    atomicAdd(&q_ws[c + 1], qy);
    atomicAdd(&q_ws[c + 2], qz);
    atomicAdd(&q_ws[c + 3], qw);
}

// Stage 2: single wave32. Each lane folds 32 columns into a partial p, then a
// v_wmma_f32_16x16x4_f32 with A = partials (A[m][0]=p[m], A[m][2]=p[m+16],
// other K-slices zero) and B = ones yields D[m][*] = p[m] + p[m+16] in the
// accumulator VGPRs. Sum the 8 D VGPRs per lane, xor-shuffle across the lane
// halves, done. EXEC is all-ones (uniform control flow) as WMMA requires.
__global__ void vl_stage2(const float* __restrict__ ws,
                          float* __restrict__ out,
                          int n) {
    const int lane = threadIdx.x;                // 0..31
    const float inv_n = 1.0f / (float)n;
    const float inv_nm1 = 1.0f / (float)(n - 1);

    float p = 0.0f;
    for (int i = lane; i < NCOLS; i += 32) {     // uniform trip count: EXEC stays full
        float S = ws[i];
        float Q = ws[NCOLS + i];
        float var = (Q - S * S * inv_n) * inv_nm1;
        p += fabsf(var - 1.0f);
    }

    v2f a; a[0] = p;    a[1] = 0.0f;             // A-matrix: 16x4, data in K=0/K=2 slices
    v2f b; b[0] = 1.0f; b[1] = 1.0f;             // B-matrix: 4x16 of ones
    v8f c = {};
    c = __builtin_amdgcn_wmma_f32_16x16x4_f32(
            /*neg_a=*/false, a, /*neg_b=*/false, b,
            /*c_mod=*/(short)0, c, /*reuse_a=*/false, /*reuse_b=*/false);

    float t = ((c[0] + c[1]) + (c[2] + c[3])) + ((c[4] + c[5]) + (c[6] + c[7]));
    t += __shfl_xor(t, 16, 32);                  // combine the two lane halves
    if (lane == 0) out[0] = t;
}

extern "C" void kernel_launch(void* const* d_in, const int* in_sizes, int n_in,
                              void* d_out, int out_size, void* d_ws, size_t ws_size,
                              hipStream_t stream) {
    const float* x = (const float*)d_in[0];
    float* out = (float*)d_out;
    float* ws = (float*)d_ws;

    const long total = (long)in_sizes[0];
    const int n = (int)(total / NCOLS);          // 65536 rows
    const int nblk = (n + ROWS_PER_BLOCK - 1) / ROWS_PER_BLOCK;  // 512 blocks

    vl_zero_ws<<<8, 256, 0, stream>>>(ws);
    vl_stage1<<<nblk, 256, 0, stream>>>(x, ws, n);
    vl_stage2<<<1, 32, 0, stream>>>(ws, out, n);
}